// FractalFeatureLayer_70824010711174
// MI455X (gfx1250) — compile-verified
//
#include <hip/hip_runtime.h>
#include <hip/hip_bf16.h>

// Petrosian fractal dimension over 5 scales of half-overlapping windows.
// x: (B=64, C=64, T=4096) f32 -> out: (B, C, 57) f32.
// Per row: s[j] = ((x[j+1]-x[j])*(x[j+2]-x[j+1]) < 0), j in [0, T-3].
// Window (start, w): zc = P[start+w-2] - P[start], P = exclusive prefix of s.
// out = L / (2L - log10(w + 0.4*zc)),  L = log10(w).

#define T_LEN   4096
#define N_WORDS 128     // T_LEN / 32 bitmask words
#define N_OUT   57      // 1 + 3 + 7 + 15 + 31

typedef float f32x4 __attribute__((ext_vector_type(4)));
typedef int   i32x4 __attribute__((ext_vector_type(4)));

__device__ __forceinline__ unsigned ballot32(bool p) {
#if __has_builtin(__builtin_amdgcn_ballot_w32)
    return __builtin_amdgcn_ballot_w32(p);
#else
    return (unsigned)__ballot(p);
#endif
}

__global__ void __launch_bounds__(128)
pfd_kernel(const float* __restrict__ x, float* __restrict__ out) {
    __shared__ float    xs[T_LEN];
    __shared__ unsigned mask[N_WORDS];
    __shared__ unsigned wpref[N_WORDS];
    __shared__ unsigned waveTot[4];

    const int t    = threadIdx.x;      // 0..127
    const int lane = t & 31;
    const int wv   = t >> 5;
    const long long row = blockIdx.x;

    const f32x4* __restrict__ g4 = (const f32x4*)(x + row * (long long)T_LEN);
    f32x4* l4 = (f32x4*)xs;

    // ---- Stage the row HBM -> LDS via the CDNA5 async data path ----
#if __has_builtin(__builtin_amdgcn_global_load_async_to_lds_b128)
    #pragma unroll
    for (int r = 0; r < 8; ++r) {
        int i = r * 128 + t;   // float4 index: 8*128*16B = 16 KB
        __builtin_amdgcn_global_load_async_to_lds_b128(
            (__attribute__((address_space(1))) i32x4*)(g4 + i),
            (__attribute__((address_space(3))) i32x4*)(l4 + i),
            /*imm offset*/0, /*cpol*/0);
    }
  #if __has_builtin(__builtin_amdgcn_s_wait_asynccnt)
    __builtin_amdgcn_s_wait_asynccnt(0);
  #else
    asm volatile("s_wait_asynccnt 0" ::: "memory");
  #endif
#else
    #pragma unroll
    for (int r = 0; r < 8; ++r) {
        int i = r * 128 + t;
        l4[i] = g4[i];
    }
#endif
    __syncthreads();

    // ---- Sign-change flags, bit-packed via wave ballot (conflict-free LDS) ----
    // Wave wv owns words [32*wv, 32*wv+32); lane l tests position j = 32*W + l.
    #pragma unroll 4
    for (int W = wv * 32; W < wv * 32 + 32; ++W) {
        int j = (W << 5) + lane;
        bool s = false;
        if (j <= T_LEN - 3) {
            float a = xs[j], b = xs[j + 1], c = xs[j + 2];
            s = (b - a) * (c - b) < 0.0f;
        }
        unsigned m = ballot32(s);
        if (lane == 0) mask[W] = m;
    }
    __syncthreads();

    // ---- Exclusive prefix of per-word popcounts (wave shuffle scan) ----
    unsigned pcv  = __popc(mask[t]);
    unsigned incl = pcv;
    #pragma unroll
    for (int d = 1; d < 32; d <<= 1) {
        unsigned n = __shfl_up(incl, d, 32);
        if (lane >= d) incl += n;
    }
    if (lane == 31) waveTot[wv] = incl;
    __syncthreads();
    unsigned off = 0;
    for (int k = 0; k < wv; ++k) off += waveTot[k];
    wpref[t] = off + incl - pcv;   // exclusive prefix of word t
    __syncthreads();

    // ---- 57 window outputs ----
    if (t < N_OUT) {
        int scale, i;
        if      (t < 1)  { scale = 0; i = t;      }
        else if (t < 4)  { scale = 1; i = t - 1;  }
        else if (t < 11) { scale = 2; i = t - 4;  }
        else if (t < 26) { scale = 3; i = t - 11; }
        else             { scale = 4; i = t - 26; }
        int w     = T_LEN >> scale;        // 4096,2048,1024,512,256
        int start = i * (w >> 1);          // multiple of 128 -> word-aligned
        int e     = start + w - 2;         // e & 31 == 30

        unsigned zc = wpref[e >> 5]
                    + __popc(mask[e >> 5] & ((1u << (e & 31)) - 1u))
                    - wpref[start >> 5];

        float fw = (float)w;
        float L  = log10f(fw);
        float denom = 2.0f * L - log10f(fw + 0.4f * (float)zc);
        out[row * N_OUT + t] = L / denom;
    }
}

extern "C" void kernel_launch(void* const* d_in, const int* in_sizes, int n_in,
                              void* d_out, int out_size, void* d_ws, size_t ws_size,
                              hipStream_t stream) {
    const float* x = (const float*)d_in[0];
    float* out = (float*)d_out;
    int rows = in_sizes[0] / T_LEN;   // B*C = 4096
    pfd_kernel<<<dim3(rows), dim3(128), 0, stream>>>(x, out);
}